// MyNet_14851996909845
// MI455X (gfx1250) — compile-verified
//
#include <hip/hip_runtime.h>
#include <hip/hip_bf16.h>

// Problem constants (match reference).
constexpr int NG = 1708;   // genes
constexpr int NH = 5;      // heads
constexpr int NC = 34;     // classes
constexpr int NB = 16;     // batch
#define LN_EPS 1e-6f
#define LOG2E  1.4426950408889634f

typedef __attribute__((ext_vector_type(2))) float v2f;
typedef __attribute__((ext_vector_type(8))) float v8f;

// ---------------- wave32 reductions -------------------------------------
__device__ __forceinline__ float waveSum(float v) {
#pragma unroll
    for (int o = 16; o > 0; o >>= 1) v += __shfl_xor(v, o, 32);
    return v;
}

// ---------------- per-(b,h) k min/max for stable softmax ----------------
// One wave per (b,h): k_j = x[b,j]*WK[h,j]; row max of q_i*k_j is then
// q_i*kmax (q>=0) or q_i*kmin (q<0).
__global__ void kminmax_kernel(const float* __restrict__ x,
                               const float* __restrict__ WK,
                               float* __restrict__ kmm) {
    const int w    = (blockIdx.x * blockDim.x + threadIdx.x) >> 5;
    const int lane = threadIdx.x & 31;
    if (w >= NB * NH) return;
    const int b = w / NH, h = w % NH;
    const float* xr = x + b * NG;
    const float* wk = WK + h * NG;
    float mn = 1e30f, mx = -1e30f;
    for (int j = lane; j < NG; j += 32) {
        const float k = xr[j] * wk[j];
        mn = fminf(mn, k);
        mx = fmaxf(mx, k);
    }
#pragma unroll
    for (int o = 16; o > 0; o >>= 1) {
        mn = fminf(mn, __shfl_xor(mn, o, 32));
        mx = fmaxf(mx, __shfl_xor(mx, o, 32));
    }
    if (lane == 0) { kmm[2 * w] = mn; kmm[2 * w + 1] = mx; }
}

// ---------------- attention (rank-1 softmax trick) ----------------------
// grid = (ceil(NG/32), NB), block = 256 (8 waves). Each wave owns 4 rows i,
// lanes stride over j. k,v for all 5 heads of this batch row live in LDS as
// interleaved float2 {k,v} (2*5*1708*4B = 68 KB dynamic; ds_load_b64 = one
// conflict-free pass over the 64 banks). Exps run as native v_exp_f32 on
// log2e-prescaled arguments.
__global__ void attn_kernel(const float* __restrict__ xin,
                            const float* __restrict__ WQ,
                            const float* __restrict__ WK,
                            const float* __restrict__ WV,
                            const float* __restrict__ W0,
                            const float* __restrict__ kmm,
                            float* __restrict__ out) {
    extern __shared__ float smem[];
    v2f* kv = (v2f*)smem;            // [NH*NG] of {k, v}

    const int b = blockIdx.y;
    const float* xr = xin + b * NG;

    for (int idx = threadIdx.x; idx < NH * NG; idx += blockDim.x) {
        const int j = idx % NG;
        const float xj = xr[j];
        v2f p; p.x = xj * WK[idx]; p.y = xj * WV[idx];
        kv[idx] = p;
    }
    __syncthreads();

    const int wave = threadIdx.x >> 5;
    const int lane = threadIdx.x & 31;
    const int ibase = (blockIdx.x * 8 + wave) * 4;
    if (ibase >= NG) return;

    int   ir[4];
    float q2[4][NH], m2[4][NH], S1[4][NH], Sv[4][NH];
#pragma unroll
    for (int r = 0; r < 4; ++r) {
        int i = ibase + r; if (i > NG - 1) i = NG - 1;   // clamped; store guarded
        ir[r] = i;
        const float xi = xr[i];
#pragma unroll
        for (int h = 0; h < NH; ++h) {
            const float qq  = xi * WQ[h * NG + i] * LOG2E;   // log2-domain
            const float kmn = kmm[(b * NH + h) * 2 + 0];
            const float kmx = kmm[(b * NH + h) * 2 + 1];
            q2[r][h] = qq;
            m2[r][h] = (qq >= 0.f) ? qq * kmx : qq * kmn;    // = log2e * rowmax
            S1[r][h] = 0.f;
            Sv[r][h] = 0.f;
        }
    }

    // Hot loop: diagonal included here, corrected after reduction.
    // Per (r,h): 1 fma + 1 native v_exp_f32 + 1 add + 1 fma.
    for (int j = lane; j < NG; j += 32) {
        float kk[NH], vv[NH];
#pragma unroll
        for (int h = 0; h < NH; ++h) {
            const v2f p = kv[h * NG + j];   // ds_load_b64
            kk[h] = p.x; vv[h] = p.y;
        }
#pragma unroll
        for (int r = 0; r < 4; ++r) {
#pragma unroll
            for (int h = 0; h < NH; ++h) {
                const float e = __builtin_amdgcn_exp2f(
                    __builtin_fmaf(q2[r][h], kk[h], -m2[r][h]));
                S1[r][h] += e;
                Sv[r][h] = __builtin_fmaf(e, vv[h], Sv[r][h]);
            }
        }
    }

#pragma unroll
    for (int r = 0; r < 4; ++r) {
        float acc = 0.f;
#pragma unroll
        for (int h = 0; h < NH; ++h) {
            const float s1 = waveSum(S1[r][h]);
            float       sv = waveSum(Sv[r][h]);
            // remove diagonal (zeroed AFTER softmax in reference; denom keeps it)
            const v2f   pd = kv[h * NG + ir[r]];
            const float ed = __builtin_amdgcn_exp2f(
                __builtin_fmaf(q2[r][h], pd.x, -m2[r][h]));
            sv -= ed * pd.y;
            acc += W0[h] * sv / s1;
        }
        if (lane == 0 && (ibase + r) < NG) out[b * NG + ibase + r] = acc;
    }
}

// ---------------- residual + layernorm (ddof=1, /(std+eps)) -------------
__global__ void lnadd_kernel(const float* __restrict__ xin,
                             const float* __restrict__ attn,
                             const float* __restrict__ ga,
                             const float* __restrict__ gb,
                             float* __restrict__ out) {
    const int b = blockIdx.x;
    const float* ar = attn + b * NG;
    float s = 0.f, sq = 0.f;
    for (int i = threadIdx.x; i < NG; i += blockDim.x) {
        const float a = ar[i];
        s += a; sq += a * a;
    }
    __shared__ float sh1[8], sh2[8], shm[2];
    s  = waveSum(s);
    sq = waveSum(sq);
    const int wave = threadIdx.x >> 5, lane = threadIdx.x & 31;
    if (lane == 0) { sh1[wave] = s; sh2[wave] = sq; }
    __syncthreads();
    if (threadIdx.x == 0) {
        float ts = 0.f, tq = 0.f;
        for (int w = 0; w < (int)(blockDim.x >> 5); ++w) { ts += sh1[w]; tq += sh2[w]; }
        const float mean = ts / (float)NG;
        float var = (tq - (float)NG * mean * mean) / (float)(NG - 1);
        var = var > 0.f ? var : 0.f;
        shm[0] = mean;
        shm[1] = 1.f / (sqrtf(var) + LN_EPS);
    }
    __syncthreads();
    const float mean = shm[0], inv = shm[1];
    for (int i = threadIdx.x; i < NG; i += blockDim.x)
        out[b * NG + i] = xin[b * NG + i] + (ga[i] * (ar[i] - mean) * inv + gb[i]);
}

// ---------------- FC (WMMA f32 16x16x4) + log_softmax -------------------
// logits[16,34] = h[16,1708] @ fc_w[34,1708]^T + b. One block, 3 waves,
// wave w owns N-tile w (N padded 34->48). K = 1708 = 427*4 exactly.
// A 16x4 f32 layout: lane L -> M = L&15; VGPR0 holds K = (L<16?0:2),
// VGPR1 holds K = (L<16?1:3)  ==> contiguous float2 per lane.
__global__ void fc_logsoftmax_kernel(const float* __restrict__ h,
                                     const float* __restrict__ fcw,
                                     const float* __restrict__ fcb,
                                     float* __restrict__ out) {
    __shared__ float slog[NB * NC];
    const int lane  = threadIdx.x & 31;
    const int ntile = threadIdx.x >> 5;
    const int hi    = lane >> 4;          // upper half-wave
    const int m     = lane & 15;          // A-row / C-column index
    const int n     = ntile * 16 + m;     // output class
    const int nsafe = (n < NC) ? n : 0;
    const int koff  = hi ? 2 : 0;

    const float* arow = h   + m * NG + koff;      // A[m][k]   = h[m, k]
    const float* brow = fcw + nsafe * NG + koff;  // B[k][n]   = fc_w[n, k]
    const v2f zero2 = (v2f)0.f;
    v8f acc = (v8f)0.f;

    for (int k0 = 0; k0 < NG; k0 += 4) {
        const v2f av = *(const v2f*)(arow + k0);
        v2f bv = *(const v2f*)(brow + k0);
        bv = (n < NC) ? bv : zero2;
        acc = __builtin_amdgcn_wmma_f32_16x16x4_f32(
            /*neg_a=*/false, av, /*neg_b=*/false, bv,
            /*c_mod=*/(short)0, acc, /*reuse_a=*/false, /*reuse_b=*/false);
    }

    const float bias = (n < NC) ? fcb[nsafe] : 0.f;
#pragma unroll
    for (int r = 0; r < 8; ++r) {
        const int mm = r + (hi ? 8 : 0);          // C row (batch index)
        if (n < NC) slog[mm * NC + n] = acc[r] + bias;
    }
    __syncthreads();

    if (threadIdx.x < NB) {
        const int row = threadIdx.x;
        float mx = -1e30f;
        for (int c = 0; c < NC; ++c) mx = fmaxf(mx, slog[row * NC + c]);
        float se = 0.f;
        for (int c = 0; c < NC; ++c) se += __expf(slog[row * NC + c] - mx);
        const float lse = mx + __logf(se);
        for (int c = 0; c < NC; ++c) out[row * NC + c] = slog[row * NC + c] - lse;
    }
}

// ---------------- launcher ----------------------------------------------
extern "C" void kernel_launch(void* const* d_in, const int* in_sizes, int n_in,
                              void* d_out, int out_size, void* d_ws, size_t ws_size,
                              hipStream_t stream) {
    const float* x    = (const float*)d_in[0];
    const float* WQ1  = (const float*)d_in[1];
    const float* WK1  = (const float*)d_in[2];
    const float* WV1  = (const float*)d_in[3];
    const float* W01  = (const float*)d_in[4];
    const float* WQ2  = (const float*)d_in[5];
    const float* WK2  = (const float*)d_in[6];
    const float* WV2  = (const float*)d_in[7];
    const float* W02  = (const float*)d_in[8];
    const float* WQ3  = (const float*)d_in[9];
    const float* WK3  = (const float*)d_in[10];
    const float* WV3  = (const float*)d_in[11];
    const float* W03  = (const float*)d_in[12];
    const float* ln_a = (const float*)d_in[13];
    const float* ln_b = (const float*)d_in[14];
    const float* fc_w = (const float*)d_in[15];
    const float* fc_b = (const float*)d_in[16];
    float* out = (float*)d_out;

    float* ws   = (float*)d_ws;
    float* kmm  = ws;                        // [NB*NH*2]
    float* attn = ws + 2 * NB * NH;          // [NB*NG]
    float* hcur = attn + NB * NG;            // [NB*NG]

    const dim3  ablk(256);
    const dim3  agrid((NG + 31) / 32, NB);   // 4 rows/wave * 8 waves = 32 rows/block
    const size_t asmem = (size_t)2 * NH * NG * sizeof(float);
    const int   mmBlocks = (NB * NH * 32 + 255) / 256;

    // Layer 1
    kminmax_kernel<<<mmBlocks, 256, 0, stream>>>(x, WK1, kmm);
    attn_kernel<<<agrid, ablk, asmem, stream>>>(x, WQ1, WK1, WV1, W01, kmm, attn);
    lnadd_kernel<<<NB, 256, 0, stream>>>(x, attn, ln_a, ln_b, hcur);
    // Layer 2 (hcur updated in place; element-wise read-before-write per thread)
    kminmax_kernel<<<mmBlocks, 256, 0, stream>>>(hcur, WK2, kmm);
    attn_kernel<<<agrid, ablk, asmem, stream>>>(hcur, WQ2, WK2, WV2, W02, kmm, attn);
    lnadd_kernel<<<NB, 256, 0, stream>>>(hcur, attn, ln_a, ln_b, hcur);
    // Layer 3
    kminmax_kernel<<<mmBlocks, 256, 0, stream>>>(hcur, WK3, kmm);
    attn_kernel<<<agrid, ablk, asmem, stream>>>(hcur, WQ3, WK3, WV3, W03, kmm, attn);
    lnadd_kernel<<<NB, 256, 0, stream>>>(hcur, attn, ln_a, ln_b, hcur);
    // FC + log_softmax (WMMA)
    fc_logsoftmax_kernel<<<1, 96, 0, stream>>>(hcur, fc_w, fc_b, out);
}